// GeometricActivation_17428977287979
// MI455X (gfx1250) — compile-verified
//
#include <hip/hip_runtime.h>

// out[b,n] = 1 / (||x_b - c_n|| / sqrt(1024) + 0.1)
// via ||x||^2 + ||c||^2 - 2 x.c, cross term as f32 WMMA GEMM (V_WMMA_F32_16X16X4_F32).
// Global->LDS staging uses gfx1250 GLOBAL_LOAD_ASYNC_TO_LDS_B128 with
// double-buffered LDS (no staging VGPRs -> no spills).

typedef __attribute__((ext_vector_type(2))) float v2f;
typedef __attribute__((ext_vector_type(8))) float v8f;
typedef __attribute__((__vector_size__(4 * sizeof(int)))) int v4i;  // matches builtin param

#define DIM    1024
#define NCENT  4096
#define NROWS  8192
#define BM     128
#define BN     128
#define KT     32
#define NCH    (DIM / KT)   // 32 K-chunks
#define LDA    36           // padded LDS row stride in floats (16B aligned, bank-conflict free)

#if __has_builtin(__builtin_amdgcn_global_load_async_to_lds_b128) && \
    __has_builtin(__builtin_amdgcn_s_wait_asynccnt)
#define HAS_ASYNC_LDS 1
#else
#define HAS_ASYNC_LDS 0
#endif

#if HAS_ASYNC_LDS
typedef __attribute__((address_space(3))) v4i* lds_v4i_ptr;
#endif

// ---------------- row sum-of-squares: one block per row -----------------
__global__ __launch_bounds__(256)
void rownorm_kernel(const float* __restrict__ m, float* __restrict__ out) {
    __shared__ float red[256];
    const int row = blockIdx.x;
    const float4 v =
        reinterpret_cast<const float4*>(m)[(size_t)row * (DIM / 4) + threadIdx.x];
    red[threadIdx.x] = v.x * v.x + v.y * v.y + v.z * v.z + v.w * v.w;
    __syncthreads();
    #pragma unroll
    for (int off = 128; off > 0; off >>= 1) {
        if ((int)threadIdx.x < off) red[threadIdx.x] += red[threadIdx.x + off];
        __syncthreads();
    }
    if (threadIdx.x == 0) out[row] = red[0];
}

// ---------------- main fused distance-GEMM kernel -----------------------
__global__ __launch_bounds__(256, 1)
void geo_act_kernel(const float* __restrict__ X, const float* __restrict__ C,
                    const float* __restrict__ xsq, const float* __restrict__ csq,
                    float* __restrict__ out) {
#if HAS_ASYNC_LDS
    __shared__ float lA[2 * BM * LDA];   // double buffered
    __shared__ float lB[2 * BN * LDA];
#else
    __shared__ float lA[BM * LDA];
    __shared__ float lB[BN * LDA];
#endif
    __shared__ float sXsq[BM];
    __shared__ float sCsq[BN];

    const int t      = threadIdx.x;
    const int blockN = blockIdx.x * BN;
    const int blockM = blockIdx.y * BM;

    const int w        = t >> 5;          // wave id 0..7
    const int lane     = t & 31;
    const int half     = lane >> 4;       // K half-select for A/B frags
    const int lr       = lane & 15;       // M index (A) / N index (B)
    const int waveMOff = (w & 3) * 32;    // 4 waves along M
    const int waveNOff = (w >> 2) * 64;   // 2 waves along N

    if (t < BM) sXsq[t]      = xsq[blockM + t];
    else        sCsq[t - BM] = csq[blockN + (t - BM)];

    v8f zero = {};
    v8f acc[2][4];
    #pragma unroll
    for (int mi = 0; mi < 2; ++mi)
        #pragma unroll
        for (int ni = 0; ni < 4; ++ni) acc[mi][ni] = zero;

#if HAS_ASYNC_LDS
    // Issue async global->LDS copies for one K-chunk into LDS buffer `buf`.
    auto issueAsync = [&](int k0, int buf) {
        #pragma unroll
        for (int i = 0; i < 4; ++i) {
            const int f   = t + i * 256;
            const int row = f >> 3;
            const int c4  = f & 7;
            const float* ga = X + (size_t)(blockM + row) * DIM + k0 + c4 * 4;
            const float* gb = C + (size_t)(blockN + row) * DIM + k0 + c4 * 4;
            float* la = &lA[buf * BM * LDA + row * LDA + c4 * 4];
            float* lb = &lB[buf * BN * LDA + row * LDA + c4 * 4];
            __builtin_amdgcn_global_load_async_to_lds_b128(
                (v4i*)ga, (lds_v4i_ptr)la, 0, 0);
            __builtin_amdgcn_global_load_async_to_lds_b128(
                (v4i*)gb, (lds_v4i_ptr)lb, 0, 0);
        }
    };

    issueAsync(0, 0);
    __builtin_amdgcn_s_wait_asynccnt(0);
    __syncthreads();
#else
    // Fallback: synchronous load -> immediate ds_store (no long-lived staging regs).
    auto fillChunk = [&](int k0) {
        #pragma unroll
        for (int i = 0; i < 4; ++i) {
            const int f   = t + i * 256;
            const int row = f >> 3;
            const int c4  = f & 7;
            float4 va = reinterpret_cast<const float4*>(
                            X + (size_t)(blockM + row) * DIM + k0)[c4];
            float4 vb = reinterpret_cast<const float4*>(
                            C + (size_t)(blockN + row) * DIM + k0)[c4];
            *reinterpret_cast<float4*>(&lA[row * LDA + c4 * 4]) = va;
            *reinterpret_cast<float4*>(&lB[row * LDA + c4 * 4]) = vb;
        }
    };
    fillChunk(0);
    __syncthreads();
#endif

    for (int chunk = 0; chunk < NCH; ++chunk) {
#if HAS_ASYNC_LDS
        const int buf = chunk & 1;
        if (chunk + 1 < NCH) issueAsync((chunk + 1) * KT, (chunk + 1) & 1);
        const int aBase = buf * BM * LDA;
        const int bBase = buf * BN * LDA;
#else
        const int aBase = 0;
        const int bBase = 0;
#endif
        // 8 K-steps of 4 over the current LDS tile
        #pragma unroll
        for (int ks = 0; ks < KT / 4; ++ks) {
            const int k = ks * 4;
            v2f a[2], b[4];
            #pragma unroll
            for (int mi = 0; mi < 2; ++mi)
                a[mi] = *reinterpret_cast<const v2f*>(
                    &lA[aBase + (waveMOff + mi * 16 + lr) * LDA + k + half * 2]);
            #pragma unroll
            for (int ni = 0; ni < 4; ++ni)
                b[ni] = *reinterpret_cast<const v2f*>(
                    &lB[bBase + (waveNOff + ni * 16 + lr) * LDA + k + half * 2]);
            #pragma unroll
            for (int mi = 0; mi < 2; ++mi)
                #pragma unroll
                for (int ni = 0; ni < 4; ++ni)
                    acc[mi][ni] = __builtin_amdgcn_wmma_f32_16x16x4_f32(
                        false, a[mi], false, b[ni],
                        (short)0, acc[mi][ni], false, false);
        }

#if HAS_ASYNC_LDS
        // Own async loads done, then workgroup-wide visibility via barrier.
        __builtin_amdgcn_s_wait_asynccnt(0);
        __syncthreads();
#else
        __syncthreads();
        if (chunk + 1 < NCH) fillChunk((chunk + 1) * KT);
        __syncthreads();
#endif
    }

    // epilogue: sq_dist -> 1/(dist/32 + 0.1)
    const float invSqrtDim = 1.0f / 32.0f;  // 1/sqrt(1024)
    #pragma unroll
    for (int mi = 0; mi < 2; ++mi) {
        #pragma unroll
        for (int ni = 0; ni < 4; ++ni) {
            const int colInB = waveNOff + ni * 16 + lr;
            const int col    = blockN + colInB;
            const float cn   = sCsq[colInB];
            #pragma unroll
            for (int j = 0; j < 8; ++j) {
                const int rowInB = waveMOff + mi * 16 + half * 8 + j;
                float sq = sXsq[rowInB] + cn - 2.0f * acc[mi][ni][j];
                sq = fmaxf(sq, 0.0f);
                const float nd = sqrtf(sq) * invSqrtDim;
                out[(size_t)(blockM + rowInB) * NCENT + col] =
                    1.0f / (nd + 0.1f);
            }
        }
    }
}

extern "C" void kernel_launch(void* const* d_in, const int* in_sizes, int n_in,
                              void* d_out, int out_size, void* d_ws, size_t ws_size,
                              hipStream_t stream) {
    const float* x = (const float*)d_in[0];   // (8192, 1024) f32
    const float* c = (const float*)d_in[1];   // (4096, 1024) f32
    float* out = (float*)d_out;               // (8192, 4096) f32
    float* xsq = (float*)d_ws;                // 8192 floats
    float* csq = xsq + NROWS;                 // 4096 floats

    rownorm_kernel<<<NROWS, 256, 0, stream>>>(x, xsq);
    rownorm_kernel<<<NCENT, 256, 0, stream>>>(c, csq);

    dim3 grid(NCENT / BN, NROWS / BM);        // (32, 64)
    geo_act_kernel<<<grid, 256, 0, stream>>>(x, c, xsq, csq, out);
}